// HeteroGraphSAGE_47665547051447
// MI455X (gfx1250) — compile-verified
//
#include <hip/hip_runtime.h>

typedef __attribute__((ext_vector_type(2))) float v2f;
typedef __attribute__((ext_vector_type(8))) float v8f;

#define LDA 132  // padded LDS row (16 rows * 132 floats)
#define TM  8    // row-tiles (16 rows each) per block in GEMM kernels

// ---- degree count (as float; exact for counts < 2^24) ----
__global__ void count_k(const int* __restrict__ dst, float* __restrict__ cnt, int E) {
  int i = blockIdx.x * blockDim.x + threadIdx.x;
  if (i < E) unsafeAtomicAdd(&cnt[dst[i]], 1.0f);
}

__global__ void recip_k(float* __restrict__ c, int n) {
  int i = blockIdx.x * blockDim.x + threadIdx.x;
  if (i < n) c[i] = 1.0f / fmaxf(c[i], 1.0f);
}

// ---- scatter mean contributions: one wave32 per edge, lane = 4-float chunk ----
__global__ void scatter_k(const float* __restrict__ X, const int* __restrict__ eSrc,
                          const int* __restrict__ eDst, const float* __restrict__ rcnt,
                          float* __restrict__ tmp, int E) {
  int gid = blockIdx.x * blockDim.x + threadIdx.x;
  int e = gid >> 5;
  int lane = gid & 31;
  if (e >= E) return;
  int s = eSrc[e];          // wave-uniform -> scalar load
  int d = eDst[e];
  float w = rcnt[d];
  float4 v = *(const float4*)(X + (size_t)s * 128 + lane * 4);
  float* p = tmp + (size_t)d * 128 + lane * 4;
  unsafeAtomicAdd(p + 0, v.x * w);
  unsafeAtomicAdd(p + 1, v.y * w);
  unsafeAtomicAdd(p + 2, v.z * w);
  unsafeAtomicAdd(p + 3, v.w * w);
}

// ---- per (layer,type) combined Wr / bl: Wsum = sum_ci coef*Wr[layer,ci] ----
__global__ void wsum_k(const float* __restrict__ Wr, const float* __restrict__ bl,
                       float* __restrict__ Wsum, float* __restrict__ bsum) {
  const int layer = blockIdx.x / 3;
  const int type  = blockIdx.x % 3;
  const float COEF[3][4] = {{0.f,1.f,0.f,0.f},   // FILE:  e_f_imp_f(ci1)
                            {1.f,1.f,2.f,1.f},   // CLASS: ci0 + ci1 + 2*ci2 + ci3
                            {1.f,2.f,2.f,2.f}};  // FUNC:  ci0 + 2*ci1 + 2*ci2 + 2*ci3
  const float* wr = Wr + (size_t)layer * 4 * 16384;
  float* ws = Wsum + (size_t)blockIdx.x * 16384;
  for (int e = threadIdx.x; e < 16384; e += blockDim.x) {
    float a = 0.f;
#pragma unroll
    for (int ci = 0; ci < 4; ++ci) a += COEF[type][ci] * wr[ci * 16384 + e];
    ws[e] = a;
  }
  if (threadIdx.x < 128) {
    const float* b = bl + (size_t)layer * 512;
    float a = 0.f;
#pragma unroll
    for (int ci = 0; ci < 4; ++ci) a += COEF[type][ci] * b[ci * 128 + threadIdx.x];
    bsum[blockIdx.x * 128 + threadIdx.x] = a;
  }
}

// ---- C[tile rows, 0:128] += A[tile rows, 0:128] @ B[128,128] ----
// 8 waves/block; wave w owns columns [16w,16w+16). B fragments register-resident
// for the whole block; LDS double-buffered with A prefetch across TM row-tiles.
__global__ void __launch_bounds__(256) gemm_acc_k(const float* __restrict__ A,
                                                  const float* __restrict__ B,
                                                  float* __restrict__ C, int ntiles) {
  __shared__ float lds[2][16 * LDA];
  const int tid  = threadIdx.x;
  const int wave = tid >> 5;
  const int lane = tid & 31;
  const int hi   = lane >> 4;      // 0: K+{0,1}/rows 0-7 ; 1: K+{2,3}/rows 8-15
  const int l15  = lane & 15;
  const int n0   = wave * 16;

  // Preload all 32 B fragments (K=128 in steps of 4): 64 VGPRs.
  v2f bf[32];
#pragma unroll
  for (int ks = 0; ks < 32; ++ks) {
    const float* bp = B + (size_t)(ks * 4 + hi * 2) * 128 + n0 + l15;
    bf[ks].x = bp[0];
    bf[ks].y = bp[128];
  }

  const int tile0 = blockIdx.x * TM;
  const int nt = (ntiles - tile0 < TM) ? (ntiles - tile0) : TM;  // >= 1, block-uniform
  const int ar = tid >> 4;          // A-load row
  const int ac = (tid & 15) * 8;    // A-load col

  float4 p0, p1;  // prefetched A tile fragment (this thread's 8 floats)
  {
    const float* s = A + ((size_t)tile0 * 16 + ar) * 128 + ac;
    p0 = *(const float4*)s;
    p1 = *(const float4*)(s + 4);
  }
  for (int tt = 0; tt < nt; ++tt) {
    float* dbuf = &lds[tt & 1][ar * LDA + ac];
    *(float4*)dbuf = p0;
    *(float4*)(dbuf + 4) = p1;
    if (tt + 1 < nt) {  // prefetch next tile while this one computes
      const float* s = A + ((size_t)(tile0 + tt + 1) * 16 + ar) * 128 + ac;
      p0 = *(const float4*)s;
      p1 = *(const float4*)(s + 4);
    }
    __syncthreads();
    const size_t m0 = (size_t)(tile0 + tt) * 16;
    float* cbase = C + (m0 + (size_t)hi * 8) * 128 + n0 + l15;
    v8f c;
#pragma unroll
    for (int i = 0; i < 8; ++i) c[i] = cbase[(size_t)i * 128];
    // preload all A fragments, then run 32 WMMAs back-to-back on the C chain
    v2f af[32];
    const float* abase = &lds[tt & 1][l15 * LDA + hi * 2];
#pragma unroll
    for (int ks = 0; ks < 32; ++ks) af[ks] = *(const v2f*)(abase + ks * 4);
#pragma unroll
    for (int ks = 0; ks < 32; ++ks)
      c = __builtin_amdgcn_wmma_f32_16x16x4_f32(false, af[ks], false, bf[ks], (short)0, c,
                                                false, false);
#pragma unroll
    for (int i = 0; i < 8; ++i) cbase[(size_t)i * 128] = c[i];
  }
}

// ---- Out = relu((Acc + H @ B + bias) * inv_nrel); row-local -> in-place safe ----
__global__ void __launch_bounds__(256) finalize_k(const float* __restrict__ Acc,
                                                  const float* __restrict__ H,
                                                  const float* __restrict__ B,
                                                  const float* __restrict__ bias,
                                                  float* __restrict__ Out, float inv_nrel,
                                                  int ntiles) {
  __shared__ float lds[2][16 * LDA];
  const int tid  = threadIdx.x;
  const int wave = tid >> 5;
  const int lane = tid & 31;
  const int hi   = lane >> 4;
  const int l15  = lane & 15;
  const int n0   = wave * 16;

  v2f bf[32];
#pragma unroll
  for (int ks = 0; ks < 32; ++ks) {
    const float* bp = B + (size_t)(ks * 4 + hi * 2) * 128 + n0 + l15;
    bf[ks].x = bp[0];
    bf[ks].y = bp[128];
  }
  const float bn = bias[n0 + l15];

  const int tile0 = blockIdx.x * TM;
  const int nt = (ntiles - tile0 < TM) ? (ntiles - tile0) : TM;
  const int ar = tid >> 4;
  const int ac = (tid & 15) * 8;

  float4 p0, p1;
  {
    const float* s = H + ((size_t)tile0 * 16 + ar) * 128 + ac;
    p0 = *(const float4*)s;
    p1 = *(const float4*)(s + 4);
  }
  for (int tt = 0; tt < nt; ++tt) {
    float* dbuf = &lds[tt & 1][ar * LDA + ac];
    *(float4*)dbuf = p0;
    *(float4*)(dbuf + 4) = p1;
    if (tt + 1 < nt) {
      const float* s = H + ((size_t)(tile0 + tt + 1) * 16 + ar) * 128 + ac;
      p0 = *(const float4*)s;
      p1 = *(const float4*)(s + 4);
    }
    __syncthreads();
    const size_t m0 = (size_t)(tile0 + tt) * 16;
    const size_t coff = (m0 + (size_t)hi * 8) * 128 + n0 + l15;
    v8f c;
#pragma unroll
    for (int i = 0; i < 8; ++i) c[i] = Acc[coff + (size_t)i * 128];
    v2f af[32];
    const float* abase = &lds[tt & 1][l15 * LDA + hi * 2];
#pragma unroll
    for (int ks = 0; ks < 32; ++ks) af[ks] = *(const v2f*)(abase + ks * 4);
#pragma unroll
    for (int ks = 0; ks < 32; ++ks)
      c = __builtin_amdgcn_wmma_f32_16x16x4_f32(false, af[ks], false, bf[ks], (short)0, c,
                                                false, false);
#pragma unroll
    for (int i = 0; i < 8; ++i)
      Out[coff + (size_t)i * 128] = fmaxf((c[i] + bn) * inv_nrel, 0.f);
  }
}

extern "C" void kernel_launch(void* const* d_in, const int* in_sizes, int n_in,
                              void* d_out, int out_size, void* d_ws, size_t ws_size,
                              hipStream_t stream) {
  const float* x[3] = {(const float*)d_in[0], (const float*)d_in[1], (const float*)d_in[2]};
  const float* Wl = (const float*)d_in[3];
  const float* bl = (const float*)d_in[4];
  const float* Wr = (const float*)d_in[5];

  // relation r -> d_in[6+r]; types: 0=FILE,1=CLASS,2=FUNCTION
  static const int RSRC[13] = {0,0,0,0,0,0, 1,1,2,2, 1,1,2};
  static const int RDST[13] = {1,2,2,0,1,2, 1,2,1,2, 2,1,2};
  static const int NTT[3] = {50000, 150000, 300000};
  static const size_t TOFF[3] = {0, 50000, 200000};
  static const float INV_NREL[3] = {1.0f, 1.0f / 5.0f, 1.0f / 7.0f};
  // groups sharing (dst_type, conv_idx) -> one Wl GEMM each
  static const int G_T[9]    = {0, 1, 1, 1, 1, 2, 2, 2, 2};
  static const int G_CI[9]   = {1, 0, 1, 2, 3, 0, 1, 2, 3};
  static const int G_N[9]    = {1, 1, 1, 2, 1, 1, 2, 2, 2};
  static const int G_R[9][2] = {{3,-1},{0,-1},{4,-1},{6,8},{11,-1},{1,-1},{2,5},{7,9},{10,12}};

  float* w    = (float*)d_ws;
  float* h1   = w;                      // 500000*128            = 64,000,000 f
  float* tmp  = w + 64000000ull;        // 300000*128            = 38,400,000 f
  float* rcnt = w + 102400000ull;       // sum of dst sizes      =  2,900,000 f
  float* wsum = w + 105300000ull;       // 2*3*128*128           =     98,304 f
  float* bsum = w + 105398304ull;       // 2*3*128               =        768 f
  float* out  = (float*)d_out;

  size_t rcOff[13], racc = 0;
  for (int r = 0; r < 13; ++r) { rcOff[r] = racc; racc += (size_t)NTT[RDST[r]]; }

  // degree reciprocals (edges are layer-invariant)
  hipMemsetAsync(rcnt, 0, racc * sizeof(float), stream);
  for (int r = 0; r < 13; ++r) {
    const int* edges = (const int*)d_in[6 + r];
    int E = in_sizes[6 + r] / 2;
    count_k<<<(E + 255) / 256, 256, 0, stream>>>(edges + E, rcnt + rcOff[r], E);
  }
  for (int r = 0; r < 13; ++r) {
    int n = NTT[RDST[r]];
    recip_k<<<(n + 255) / 256, 256, 0, stream>>>(rcnt + rcOff[r], n);
  }
  wsum_k<<<6, 256, 0, stream>>>(Wr, bl, wsum, bsum);

  for (int layer = 0; layer < 2; ++layer) {
    hipMemsetAsync(out, 0, (size_t)500000 * 128 * sizeof(float), stream);
    for (int g = 0; g < 9; ++g) {
      const int t = G_T[g], ci = G_CI[g];
      const int nd = NTT[t];
      hipMemsetAsync(tmp, 0, (size_t)nd * 128 * sizeof(float), stream);
      for (int j = 0; j < G_N[g]; ++j) {
        const int r = G_R[g][j];
        const int* edges = (const int*)d_in[6 + r];
        const int E = in_sizes[6 + r] / 2;
        const float* src = (layer == 0) ? x[RSRC[r]] : (h1 + TOFF[RSRC[r]] * 128);
        const long threads = (long)E * 32;
        scatter_k<<<(unsigned)((threads + 255) / 256), 256, 0, stream>>>(
            src, edges, edges + E, rcnt + rcOff[r], tmp, E);
      }
      const int ntiles = nd / 16;
      gemm_acc_k<<<(ntiles + TM - 1) / TM, 256, 0, stream>>>(
          tmp, Wl + ((size_t)layer * 4 + ci) * 16384, out + TOFF[t] * 128, ntiles);
    }
    for (int t = 0; t < 3; ++t) {
      const float* H = (layer == 0) ? x[t] : (h1 + TOFF[t] * 128);
      float* dst = (layer == 0) ? (h1 + TOFF[t] * 128) : (out + TOFF[t] * 128);
      const int ntiles = NTT[t] / 16;
      finalize_k<<<(ntiles + TM - 1) / TM, 256, 0, stream>>>(
          out + TOFF[t] * 128, H,
          wsum + ((size_t)layer * 3 + t) * 16384,
          bsum + ((size_t)layer * 3 + t) * 128,
          dst, INV_NREL[t], ntiles);
    }
  }
}